// GraphAttentionPooling_78967268704508
// MI455X (gfx1250) — compile-verified
//
#include <hip/hip_runtime.h>
#include <hip/hip_bf16.h>
#include <stdint.h>

// GraphAttentionPooling for MI455X (gfx1250), compile-only tuned.
// B=32, N=3072, F=256, P=3 -> 32768 independent windows of 3x256 floats.
// Memory-bound (128 MiB @ 23.3 TB/s ~ 5.7us floor): use CDNA5 async
// global->LDS loads (ASYNCcnt) with per-wave double buffering, NT cache
// policy on the single-use streams, wave32 shuffle reduction for the dots.

typedef float v4f __attribute__((ext_vector_type(4)));  // native vector (not HIP_vector_type)

namespace {
constexpr int kF            = 256;           // features
constexpr int kP            = 3;             // pooling window
constexpr int kWinFloats    = kF * kP;       // 768 floats / window
constexpr int kWinBytes     = kWinFloats * 4;// 3072 bytes / window
constexpr int kWavesPerBlk  = 8;
constexpr int kTPB          = kWavesPerBlk * 32;
constexpr int kWPW          = 4;             // windows per wave (double-buffered)
}

// Issue one full window (3 KB) as 6 async B128 copies: lane l moves bytes
// [l*16 + k*512, +16). The same IOFFSET is applied to BOTH the LDS dest and
// the global source (CDNA5 ISA pseudocode), so a single LDS-base
// VGPR and a single 32-bit global-offset VGPR cover all 6 chunks. Input is
// single-use -> non-temporal hint keeps it from displacing cache state.
__device__ __forceinline__ void async_copy_window(unsigned lds_off, unsigned gvoff,
                                                  const float* __restrict__ base) {
  asm volatile("global_load_async_to_lds_b128 %0, %1, %2 offset:0 th:TH_LOAD_NT"    :: "v"(lds_off), "v"(gvoff), "s"(base) : "memory");
  asm volatile("global_load_async_to_lds_b128 %0, %1, %2 offset:512 th:TH_LOAD_NT"  :: "v"(lds_off), "v"(gvoff), "s"(base) : "memory");
  asm volatile("global_load_async_to_lds_b128 %0, %1, %2 offset:1024 th:TH_LOAD_NT" :: "v"(lds_off), "v"(gvoff), "s"(base) : "memory");
  asm volatile("global_load_async_to_lds_b128 %0, %1, %2 offset:1536 th:TH_LOAD_NT" :: "v"(lds_off), "v"(gvoff), "s"(base) : "memory");
  asm volatile("global_load_async_to_lds_b128 %0, %1, %2 offset:2048 th:TH_LOAD_NT" :: "v"(lds_off), "v"(gvoff), "s"(base) : "memory");
  asm volatile("global_load_async_to_lds_b128 %0, %1, %2 offset:2560 th:TH_LOAD_NT" :: "v"(lds_off), "v"(gvoff), "s"(base) : "memory");
}

__device__ __forceinline__ float dot8(const v4f& a0, const v4f& a1,
                                      const v4f& b0, const v4f& b1) {
  return a0.x * b0.x + a0.y * b0.y + a0.z * b0.z + a0.w * b0.w +
         a1.x * b1.x + a1.y * b1.y + a1.z * b1.z + a1.w * b1.w;
}

__global__ __launch_bounds__(kTPB)
void GraphAttentionPooling_78967268704508_kernel(const float* __restrict__ x,
                                                 const float* __restrict__ Wv,
                                                 const float* __restrict__ Wb,
                                                 float* __restrict__ out,
                                                 int nwin) {
  // Per-wave double buffer: 8 waves * 2 * 768 floats = 48 KB of the 320 KB WGP LDS
  // (leaves room for ~6 resident blocks per WGP).
  __shared__ float smem[kWavesPerBlk * 2 * kWinFloats];

  const int lane = threadIdx.x & 31;
  const int wave = threadIdx.x >> 5;
  const int gw   = blockIdx.x * kWavesPerBlk + wave;
  const int w0   = gw * kWPW;
  if (w0 + kWPW > nwin) return;  // launch config makes this exact; safety only

  // 32-bit wave-relative LDS byte offsets for the two buffers (lane base = lane*16B).
  const unsigned lds0 =
      (unsigned)(uintptr_t)&smem[(wave * 2 + 0) * kWinFloats + lane * 4];
  const unsigned lds1 =
      (unsigned)(uintptr_t)&smem[(wave * 2 + 1) * kWinFloats + lane * 4];

  // Lane-private slice of the weight vector (8 floats), loaded once, reused.
  const v4f* wp = reinterpret_cast<const v4f*>(Wv) + lane * 2;
  const v4f w4a = wp[0];
  const v4f w4b = wp[1];
  const float fbias = Wb[0];

  // Prime the pipeline: window w0 -> buffer 0.
  async_copy_window(lds0, (unsigned)w0 * kWinBytes + (unsigned)lane * 16u, x);

#pragma unroll
  for (int i = 0; i < kWPW; ++i) {
    const int w = w0 + i;
    if (i + 1 < kWPW) {
      // The buffer we are about to overwrite was read via DS ops last
      // iteration; async LDS writes are NOT ordered against DS ops
      // (CDNA5 ISA), so drain DScnt before reusing the buffer.
      asm volatile("s_wait_dscnt 0x0" ::: "memory");
      // Prefetch next window into the other buffer, then wait until only
      // those 6 async ops remain outstanding (async loads complete in order).
      async_copy_window(((i + 1) & 1) ? lds1 : lds0,
                        (unsigned)(w + 1) * kWinBytes + (unsigned)lane * 16u, x);
      asm volatile("s_wait_asynccnt 0x6" ::: "memory");
    } else {
      asm volatile("s_wait_asynccnt 0x0" ::: "memory");
    }

    const int base = (wave * 2 + (i & 1)) * kWinFloats + lane * 8;
    const v4f x0a = *reinterpret_cast<const v4f*>(&smem[base + 0 * kF]);
    const v4f x0b = *reinterpret_cast<const v4f*>(&smem[base + 0 * kF + 4]);
    const v4f x1a = *reinterpret_cast<const v4f*>(&smem[base + 1 * kF]);
    const v4f x1b = *reinterpret_cast<const v4f*>(&smem[base + 1 * kF + 4]);
    const v4f x2a = *reinterpret_cast<const v4f*>(&smem[base + 2 * kF]);
    const v4f x2b = *reinterpret_cast<const v4f*>(&smem[base + 2 * kF + 4]);

    // Partial dots over this lane's 8 features, then wave32 tree reduction.
    float d0 = dot8(x0a, x0b, w4a, w4b);
    float d1 = dot8(x1a, x1b, w4a, w4b);
    float d2 = dot8(x2a, x2b, w4a, w4b);
#pragma unroll
    for (int off = 16; off > 0; off >>= 1) {
      d0 += __shfl_xor(d0, off, 32);
      d1 += __shfl_xor(d1, off, 32);
      d2 += __shfl_xor(d2, off, 32);
    }

    // Stable 3-way softmax (identical in every lane).
    const float s0 = d0 + fbias, s1 = d1 + fbias, s2 = d2 + fbias;
    const float m  = fmaxf(s0, fmaxf(s1, s2));
    const float e0 = __expf(s0 - m), e1 = __expf(s1 - m), e2 = __expf(s2 - m);
    const float inv = 1.0f / (e0 + e1 + e2);
    const float a0 = e0 * inv, a1 = e1 * inv, a2 = e2 * inv;

    v4f oa, ob;
    oa.x = a0 * x0a.x + a1 * x1a.x + a2 * x2a.x;
    oa.y = a0 * x0a.y + a1 * x1a.y + a2 * x2a.y;
    oa.z = a0 * x0a.z + a1 * x1a.z + a2 * x2a.z;
    oa.w = a0 * x0a.w + a1 * x1a.w + a2 * x2a.w;
    ob.x = a0 * x0b.x + a1 * x1b.x + a2 * x2b.x;
    ob.y = a0 * x0b.y + a1 * x1b.y + a2 * x2b.y;
    ob.z = a0 * x0b.z + a1 * x1b.z + a2 * x2b.z;
    ob.w = a0 * x0b.w + a1 * x1b.w + a2 * x2b.w;

    // Output is write-once -> non-temporal stores (th:TH_STORE_NT).
    v4f* op = reinterpret_cast<v4f*>(out + (size_t)w * kF) + lane * 2;
    __builtin_nontemporal_store(oa, op + 0);
    __builtin_nontemporal_store(ob, op + 1);
  }
}

extern "C" void kernel_launch(void* const* d_in, const int* in_sizes, int n_in,
                              void* d_out, int out_size, void* d_ws, size_t ws_size,
                              hipStream_t stream) {
  const float* x  = (const float*)d_in[0];  // [B, N, F] fp32
  const float* Wv = (const float*)d_in[1];  // [F] fp32
  const float* Wb = (const float*)d_in[2];  // [1] fp32
  float* out = (float*)d_out;               // [B, S, F, 1] fp32

  const int nwin   = in_sizes[0] / kWinFloats;            // 32768
  const int blocks = nwin / (kWavesPerBlk * kWPW);        // 1024

  GraphAttentionPooling_78967268704508_kernel<<<blocks, kTPB, 0, stream>>>(
      x, Wv, Wb, out, nwin);
}